// WaveletTransform_14396730376549
// MI455X (gfx1250) — compile-verified
//
#include <hip/hip_runtime.h>

// Haar (db1) single-level DWT along rows, f32, 8192x4096.
// Memory-bound (0.25 flop/byte): floor = 256 MiB / 23.3 TB/s ~= 11.5 us.
// Butterfly expressed as D = X_tile(16x16) * W(16x16) via 4 chained
// V_WMMA_F32_16X16X4_F32 (f32 precision preserved). Each wave handles 4
// column tiles (8 KB of stream) so the constant-W build and address setup
// amortize: kernel must stay issue-rate-limited *below* the 23.3 TB/s
// HBM requirement (~73 B/WGP-cycle). NT hints keep the 256 MiB stream
// from thrashing the 192 MB L2.

typedef float v2f __attribute__((ext_vector_type(2)));
typedef float v4f __attribute__((ext_vector_type(4)));
typedef float v8f __attribute__((ext_vector_type(8)));

#define INV_SQRT2 0.70710678118654752440f
#define FDIM 4096              // compile-time row length: shift-based addressing
#define LDS_STRIDE 20          // dwords: conflict-free column reads, 16B-aligned rows
#define WAVES_PER_BLOCK 8
#define NTILES 4               // 16-col tiles per wave

// W[k][n]: Haar butterfly matrix. Columns 0..7 -> cA, 8..15 -> cD.
__device__ __forceinline__ float wval(int k, int n) {
  const float c = INV_SQRT2;
  int p = k >> 1;
  float s = (k & 1) ? -c : c;
  return (n == p) ? c : ((n == (p + 8)) ? s : 0.0f);
}

__global__ __launch_bounds__(256) void haar_dwt_wmma(const float* __restrict__ x,
                                                     float* __restrict__ out) {
  __shared__ float lds[WAVES_PER_BLOCK][NTILES][16 * LDS_STRIDE];

  const int lane = threadIdx.x & 31;
  const int wave = threadIdx.x >> 5;

  const int row0 = blockIdx.y * 16;
  const int col0 = (blockIdx.x * WAVES_PER_BLOCK + wave) * (16 * NTILES);

  // ---- global -> LDS, coalesced: 4 tiles of 16x16 floats, 2x b128/lane/tile.
  //      All 8 loads issued up front for memory-level parallelism. ----
  {
    const int r  = lane >> 1;        // 0..15
    const int cb = (lane & 1) * 8;   // 0 or 8
    const float* src = x + ((size_t)(row0 + r) << 12) + col0 + cb;   // FDIM = 4096
    float* dst = &lds[wave][0][r * LDS_STRIDE + cb];                 // 80r+{0,32}B: 16B aligned
#pragma unroll
    for (int t = 0; t < NTILES; ++t) {
      v4f d0 = __builtin_nontemporal_load((const v4f*)(src + t * 16));
      v4f d1 = __builtin_nontemporal_load((const v4f*)(src + t * 16 + 4));
      *(v4f*)(dst + t * 16 * LDS_STRIDE)     = d0;
      *(v4f*)(dst + t * 16 * LDS_STRIDE + 4) = d1;
    }
  }

  const int n  = lane & 15;          // N column (and tile-row for A reads)
  const int h  = lane >> 4;          // half-wave
  const int kb = 2 * h;              // K base within a 4-wide chunk (A & B layouts)

  // ---- constant B operand (4x16 slices of W), built once per wave.
  //      B layout: lanes 0-15 hold K = {0,1}, lanes 16-31 hold K = {2,3}. ----
  v2f bmat[4];
#pragma unroll
  for (int j = 0; j < 4; ++j) {
    int k0 = 4 * j + kb;
    bmat[j].x = wval(k0,     n);
    bmat[j].y = wval(k0 + 1, n);
  }

  const size_t orow = ((size_t)(row0 + n) << 12) + h * (FDIM / 2) + (col0 >> 1);

#pragma unroll
  for (int t = 0; t < NTILES; ++t) {
    float* tile = lds[wave][t];      // wave-private slice: no barriers needed

    // ---- D = X * W via 4 chained V_WMMA_F32_16X16X4_F32.
    // A layout (16x4 f32): lane m%16 row m; VGPR0/1 = K {0,1} (lanes<16) or {2,3}.
    v8f acc = {};
#pragma unroll
    for (int j = 0; j < 4; ++j) {
      v2f a = *(const v2f*)&tile[n * LDS_STRIDE + 4 * j + kb];  // 8B aligned
      acc = __builtin_amdgcn_wmma_f32_16x16x4_f32(false, a, false, bmat[j],
                                                  (short)0, acc, false, false);
    }

    // ---- D scatter to LDS (banks (20m+n)%64 conflict-free), then coalesced
    //      b128 NT stores: h=0 lanes write the cA half, h=1 lanes the cD half ----
#pragma unroll
    for (int v = 0; v < 8; ++v)
      tile[(v + 8 * h) * LDS_STRIDE + n] = acc[v];   // D[m=v+8h][n]

    float* dstp = out + orow + t * 8;
    v4f s0 = *(const v4f*)&tile[n * LDS_STRIDE + 8 * h];       // D[n][8h..8h+3]
    v4f s1 = *(const v4f*)&tile[n * LDS_STRIDE + 8 * h + 4];   // D[n][8h+4..8h+7]
    __builtin_nontemporal_store(s0, (v4f*)dstp);
    __builtin_nontemporal_store(s1, (v4f*)(dstp + 4));
  }
}

extern "C" void kernel_launch(void* const* d_in, const int* in_sizes, int n_in,
                              void* d_out, int out_size, void* d_ws, size_t ws_size,
                              hipStream_t stream) {
  (void)n_in; (void)out_size; (void)d_ws; (void)ws_size;
  const float* x = (const float*)d_in[0];
  float* out = (float*)d_out;

  const int nrows = in_sizes[0] / FDIM;    // 8192

  dim3 block(256);                                                   // 8 wave32
  dim3 grid(FDIM / (16 * WAVES_PER_BLOCK * NTILES), nrows / 16);     // (8, 512)
  hipLaunchKernelGGL(haar_dwt_wmma, grid, block, 0, stream, x, out);
}